// CLloss_180388627321
// MI455X (gfx1250) — compile-verified
//
#include <hip/hip_runtime.h>
#include <hip/hip_bf16.h>

typedef __attribute__((ext_vector_type(2))) float v2f;
typedef __attribute__((ext_vector_type(4))) float v4f;
typedef __attribute__((ext_vector_type(8))) float v8f;

#define BB 8192   // batch
#define DD 2048   // dim
constexpr int KC     = 64;        // K-chunk staged in LDS
constexpr int RPB    = 128;       // rows per block (8 waves x 16-row WMMA tile)
constexpr int STRIDE = KC + 4;    // 68 words: 4r+{0..3} mod 64 -> conflict-free B64 reads

// ---------------- Kernel A: anchor = normalize(embed[0]); na = max(||a||, 1e-6)
__global__ void __launch_bounds__(256) CL_anchor(const float* __restrict__ embed,
                                                 float* __restrict__ anc) {
    __shared__ float red[256];
    const int t = threadIdx.x;
    float vals[DD / 256];
    float ss = 0.0f;
#pragma unroll
    for (int i = 0; i < DD / 256; ++i) {
        float v = embed[t + 256 * i];
        vals[i] = v;
        ss += v * v;
    }
    red[t] = ss;
    __syncthreads();
    for (int s = 128; s > 0; s >>= 1) {
        if (t < s) red[t] += red[t + s];
        __syncthreads();
    }
    const float norm  = sqrtf(red[0]);
    const float denom = fmaxf(norm, 1e-12f);
#pragma unroll
    for (int i = 0; i < DD / 256; ++i) anc[t + 256 * i] = vals[i] / denom;
    if (t == 0) anc[DD] = fmaxf(norm / denom, 1e-6f);  // na = max(||a||, COS_EPS)
}

// ---------------- Kernel B: neg[j] = -(ee[j].a) / (na * max(||ee[j]||,1e-6)) / T
// GEMV done with V_WMMA_F32_16X16X4_F32 (anchor broadcast into all 16 B columns),
// row sum-of-squares fused in the same LDS-staged pass.
__global__ void __launch_bounds__(256) CL_gemv_wmma(const float* __restrict__ ee,
                                                    const float* __restrict__ anc,
                                                    float* __restrict__ negOut) {
    __shared__ float As[RPB * STRIDE];   // 34,816 B
    __shared__ float Anc[DD];            //  8,192 B
    __shared__ float DotS[RPB];          //    512 B

    const int t       = threadIdx.x;
    const int lane    = t & 31;
    const int wave    = t >> 5;          // 0..7
    const int rowBase = blockIdx.x * RPB;
    const int r       = lane & 15;       // row within 16-row tile
    const int koff    = (lane >> 4) << 1;// ISA A-layout: lanes 0-15 -> K0/K1, 16-31 -> K2/K3
    const int tile0   = wave * 16;

    for (int i = t; i < DD; i += 256) Anc[i] = anc[i];
    const float na = anc[DD];

    v8f acc = {};
    float ss = 0.0f;

    for (int k0 = 0; k0 < DD; k0 += KC) {
        __syncthreads();   // previous chunk consumed (also covers Anc staging)
        // Stage 128 rows x 64 cols, float4 coalesced: 256B contiguous per row.
#pragma unroll
        for (int i = 0; i < 8; ++i) {
            int idx = t + 256 * i;       // 0..2047 float4 slots
            int row = idx >> 4;
            int c4  = idx & 15;
            v4f v = *(const v4f*)&ee[(size_t)(rowBase + row) * DD + k0 + c4 * 4];
            *(v4f*)&As[row * STRIDE + c4 * 4] = v;
        }
        __syncthreads();

        const float* Abase = &As[(tile0 + r) * STRIDE + koff];
        const float* Bbase = &Anc[k0 + koff];
#pragma unroll
        for (int ks = 0; ks < KC; ks += 4) {
            v2f av = *(const v2f*)&Abase[ks];   // A frag (conflict-free B64)
            v2f bv = *(const v2f*)&Bbase[ks];   // B frag (LDS broadcast)
            ss += av.x * av.x + av.y * av.y;    // row sumsq (halves cover all K)
            acc = __builtin_amdgcn_wmma_f32_16x16x4_f32(
                false, av, false, bv, (short)0, acc, false, false);
        }
    }

    // sum-of-squares: lane L holds half of row (L&15); fold halves
    const float ssTot = ss + __shfl_xor(ss, 16, 32);

    // D layout: lane n<16 holds D[i][n] in acc[i]; lane 16+n holds D[8+i][n]
    if (lane == 0) {
#pragma unroll
        for (int i = 0; i < 8; ++i) DotS[tile0 + i] = acc[i];
    }
    if (lane == 16) {
#pragma unroll
        for (int i = 0; i < 8; ++i) DotS[tile0 + 8 + i] = acc[i];
    }
    __syncthreads();

    if (lane < 16) {
        const float dot  = DotS[tile0 + r];
        const float nb   = fmaxf(sqrtf(ssTot), 1e-6f);
        const float cosv = dot / (na * nb);
        negOut[rowBase + tile0 + r] = -cosv * 10.0f;   // -cos / T, T = 0.1
    }
}

// ---------------- Kernel C: scalar loss reductions (j != 0 masked)
__global__ void __launch_bounds__(256) CL_finalize(const float* __restrict__ neg,
                                                   const float* __restrict__ labels,
                                                   float* __restrict__ out) {
    __shared__ float redA[256];
    __shared__ float redB[256];
    const int t = threadIdx.x;
    const float l0 = labels[0];

    float esum = 0.0f, csum = 0.0f;
    for (int j = t; j < BB; j += 256) {
        if (j != 0) {
            esum += expf(neg[j]);
            csum += l0 * labels[j];
        }
    }
    redA[t] = esum; redB[t] = csum;
    __syncthreads();
    for (int s = 128; s > 0; s >>= 1) {
        if (t < s) { redA[t] += redA[t + s]; redB[t] += redB[t + s]; }
        __syncthreads();
    }
    const float E0    = 1e-12f + redA[0];
    const float C0    = 1e-12f + redB[0];
    const float logE0 = logf(E0);
    __syncthreads();

    float lsum = 0.0f;
    for (int j = t; j < BB; j += 256) {
        if (j != 0) lsum += -(l0 * labels[j] / C0) * (neg[j] - logE0);
    }
    redA[t] = lsum;
    __syncthreads();
    for (int s = 128; s > 0; s >>= 1) {
        if (t < s) redA[t] += redA[t + s];
        __syncthreads();
    }
    if (t == 0) out[0] = redA[0] / (float)BB;
}

extern "C" void kernel_launch(void* const* d_in, const int* in_sizes, int n_in,
                              void* d_out, int out_size, void* d_ws, size_t ws_size,
                              hipStream_t stream) {
    const float* embed  = (const float*)d_in[0];   // [8192, 2048]
    const float* ee     = (const float*)d_in[1];   // [8192, 2048]
    const float* labels = (const float*)d_in[2];   // [8192]

    float* ws  = (float*)d_ws;
    float* anc = ws;           // [0, 2048) anchor, [2048] na
    float* neg = ws + 2560;    // [2560, 2560+8192) per-row neg values

    CL_anchor   <<<1,        256, 0, stream>>>(embed, anc);
    CL_gemv_wmma<<<BB / RPB, 256, 0, stream>>>(ee, anc, neg);
    CL_finalize <<<1,        256, 0, stream>>>(neg, labels, (float*)d_out);
}